// Individual_20263655702845
// MI455X (gfx1250) — compile-verified
//
#include <hip/hip_runtime.h>
#include <math.h>

#define B_TOTAL 16384
#define IN_DIM  512
#define MID     1280
#define NODES   1792
#define OUT_DIM 256
#define CHUNKS  80           // MID / 16

typedef __attribute__((ext_vector_type(2))) float v2f;
typedef __attribute__((ext_vector_type(8))) float v8f;

// LDS row strides (padded +2 floats -> bank-conflict-free 16-row column access)
#define XS_STRIDE 514
#define MS_STRIDE 1282

// ---------------------------------------------------------------------------
// Init kernel: effW^T[m][n] = weight[n][m] * connectivity[n][m]
// Transposed so WMMA B-fragments (K = 2*hi+v contiguous) are one v2f load.
// ---------------------------------------------------------------------------
__global__ void build_eff_wt(const float* __restrict__ w,
                             const int* __restrict__ conn,
                             float* __restrict__ wt) {
    int idx = blockIdx.x * blockDim.x + threadIdx.x;   // over MID*NODES, m-major
    if (idx >= MID * NODES) return;
    int m = idx / NODES;
    int n = idx - m * NODES;
    wt[idx] = w[n * MID + m] * (float)conn[n * MID + m];
}

// ---------------------------------------------------------------------------
// Main kernel: one 16-row batch tile per workgroup (128 threads = 4 wave32s).
// For each 16-column chunk c:
//   acc[16x16] = x_tile @ effW[0:512, chunk]           (WMMA f32 16x16x4)
//              + res_tile @ effW[512:512+16c, chunk]   (WMMA f32 16x16x4)
//   then 16-step triangular solve with sigmoid/bias/existence (strictly
//   lower-triangular within the chunk: the diagonal slot is zero before its
//   own step, exactly as in the reference).
// ---------------------------------------------------------------------------
__global__ __launch_bounds__(128) void topo_forward(
        const float* __restrict__ x,
        const float* __restrict__ wt,      // effW^T [MID][NODES]
        const float* __restrict__ bias,
        const int*   __restrict__ exist,
        float* __restrict__ out) {
    extern __shared__ char smem_raw[];
    float* xs  = (float*)smem_raw;                 // [16][XS_STRIDE]   x tile
    float* Ms  = xs  + 16 * XS_STRIDE;             // [16][MS_STRIDE]   mid results
    float* red = Ms  + 16 * MS_STRIDE;             // [4][16][16]       per-wave partials
    float* Wc  = red + 4 * 16 * 16;                // [16][16]          in-chunk weights
    float* Sol = Wc  + 16 * 16;                    // [16][17]          solve tile

    const int tid  = threadIdx.x;
    const int lane = tid & 31;
    const int wv   = tid >> 5;        // wave id 0..3
    const int lo   = lane & 15;       // M (A) / N (B,C,D)
    const int hi   = lane >> 4;       // K-half selector
    const int row0 = blockIdx.x * 16;

    // Stage the 16x512 x tile (coalesced)
    for (int idx = tid; idx < 16 * IN_DIM; idx += 128) {
        int r = idx >> 9;
        int k = idx & 511;
        xs[r * XS_STRIDE + k] = x[(size_t)(row0 + r) * IN_DIM + k];
    }
    __syncthreads();

    for (int c = 0; c < CHUNKS; ++c) {
        const int col0 = c * 16;
        const float* wcol = wt + (size_t)(col0 + lo) * NODES;  // B column base (N = lo)
        v8f acc = {0.f,0.f,0.f,0.f,0.f,0.f,0.f,0.f};

        // --- x contribution: K = 512, contiguous quarter per wave -----------
        {
            const float* xrow = xs + lo * XS_STRIDE + 2 * hi;  // A: M=lo, K=2*hi+v
            const float* wp   = wcol + 2 * hi;                 // B: N=lo, K=2*hi+v
            const int kend = wv * 128 + 128;
            for (int k = wv * 128; k < kend; k += 4) {
                v2f a = *(const v2f*)(xrow + k);
                v2f b = *(const v2f*)(wp + k);
                acc = __builtin_amdgcn_wmma_f32_16x16x4_f32(
                        false, a, false, b, (short)0, acc, false, false);
            }
        }
        // --- mid contribution: K = 16*c, K-steps interleaved over waves -----
        {
            const float* mrow = Ms + lo * MS_STRIDE + 2 * hi;
            const float* wp   = wcol + IN_DIM + 2 * hi;
            for (int s = wv; s < 4 * c; s += 4) {
                int k = 4 * s;
                v2f a = *(const v2f*)(mrow + k);
                v2f b = *(const v2f*)(wp + k);
                acc = __builtin_amdgcn_wmma_f32_16x16x4_f32(
                        false, a, false, b, (short)0, acc, false, false);
            }
        }
        // --- dump partial accumulators (C layout: M = r + 8*hi, N = lo) ----
        {
            float* rp = red + wv * 256 + hi * 128 + lo;
            #pragma unroll
            for (int r = 0; r < 8; ++r)
                rp[r * 16] = acc[r];
        }
        __syncthreads();

        if (wv == 0) {          // reduce 4 partials -> Sol
            for (int idx = lane; idx < 256; idx += 32)
                Sol[(idx >> 4) * 17 + (idx & 15)] =
                    red[idx] + red[256 + idx] + red[512 + idx] + red[768 + idx];
        } else if (wv == 1) {   // fetch in-chunk 16x16 weight block
            for (int idx = lane; idx < 256; idx += 32) {
                int n = idx >> 4, n2 = idx & 15;
                Wc[n * 16 + n2] =
                    wt[(size_t)(col0 + n2) * NODES + IN_DIM + col0 + n];
            }
        }
        __syncthreads();

        // --- triangular micro-solve: lane m owns batch row m ---------------
        if (wv == 0 && lane < 16) {
            const int m = lane;
            for (int n = 0; n < 16; ++n) {
                const int i = col0 + n;
                float d = Sol[m * 17 + n];
                float r = (1.0f / (1.0f + expf(-d)) + bias[i]) * (float)exist[i];
                Ms[m * MS_STRIDE + col0 + n] = r;
                for (int n2 = n + 1; n2 < 16; ++n2)
                    Sol[m * 17 + n2] += r * Wc[n * 16 + n2];
            }
        }
        __syncthreads();
    }

    // Emit last 256 mid columns (coalesced)
    for (int idx = tid; idx < 16 * OUT_DIM; idx += 128) {
        int r = idx >> 8;
        int cc = idx & 255;
        out[(size_t)(row0 + r) * OUT_DIM + cc] =
            Ms[r * MS_STRIDE + (MID - OUT_DIM) + cc];
    }
}

// ---------------------------------------------------------------------------
extern "C" void kernel_launch(void* const* d_in, const int* in_sizes, int n_in,
                              void* d_out, int out_size, void* d_ws, size_t ws_size,
                              hipStream_t stream) {
    const float* x     = (const float*)d_in[0];   // [16384, 512]
    const float* w     = (const float*)d_in[1];   // [1792, 1280]
    const float* bias  = (const float*)d_in[2];   // [1280]
    const int*   conn  = (const int*)  d_in[3];   // [1792, 1280]
    const int*   exist = (const int*)  d_in[4];   // [1280]
    float* out = (float*)d_out;                   // [16384, 256]
    float* wt  = (float*)d_ws;                    // effW^T: MID*NODES floats (~8.75 MB)

    build_eff_wt<<<(MID * NODES + 255) / 256, 256, 0, stream>>>(w, conn, wt);

    size_t smem = (size_t)(16 * XS_STRIDE + 16 * MS_STRIDE +
                           4 * 16 * 16 + 16 * 16 + 16 * 17) * sizeof(float);
    topo_forward<<<B_TOTAL / 16, 128, smem, stream>>>(x, wt, bias, exist, out);
}